// NeighborPoolingLayer_15350213116604
// MI455X (gfx1250) — compile-verified
//
#include <hip/hip_runtime.h>

typedef __attribute__((ext_vector_type(2))) float v2f;
typedef __attribute__((ext_vector_type(8))) float v8f;

#define C_CHANNELS 128
#define ROWS_PER_BLOCK 16   // 16 output segments per workgroup
#define CHUNK 256           // edges staged in LDS per pass (one per thread)

// Segment-mean via one-hot WMMA:
//   D[16 rows x 16 chans] += A[16x4 membership] * B[4 edges x 16 chans]
// 8 waves per block, wave w owns channels [16w, 16w+16).
__global__ __launch_bounds__(256) void neighbor_pool_wmma(
    const float* __restrict__ feat,     // [N, 128]
    const int*   __restrict__ nidx,     // [E]
    const int*   __restrict__ splits,   // [M+1]
    float*       __restrict__ out,      // [M, 128]
    int M)
{
    __shared__ int   s_split[ROWS_PER_BLOCK + 1];
    __shared__ float s_inv[ROWS_PER_BLOCK];
    __shared__ int   s_idx[CHUNK];

    const int tid  = threadIdx.x;
    const int lane = tid & 31;
    const int wave = tid >> 5;                 // 0..7
    const int base = blockIdx.x * ROWS_PER_BLOCK;

    // 17 CSR boundaries for this block (clamped for the tail block:
    // dummy rows get lo==hi==E -> membership bits always 0).
    if (tid <= ROWS_PER_BLOCK) {
        int r = base + tid;
        if (r > M) r = M;
        s_split[tid] = splits[r];
    }
    __syncthreads();

    const int      r     = lane & 15;          // A-row / B-col within tile
    const int      myLo  = s_split[r];         // this lane's segment edge range
    const unsigned myCnt = (unsigned)(s_split[r + 1] - myLo);
    const int      s0    = s_split[0];
    const int      s1    = s_split[ROWS_PER_BLOCK];
    const int      k0    = (lane >= 16) ? 2 : 0;    // K-slot pair of lane half
    const unsigned chan  = (unsigned)(wave * 16 + r);  // this lane's channel

    v8f acc = {};                              // 16x16 f32 accumulator tile

    for (int cbeg = s0; cbeg < s1; cbeg += CHUNK) {
        int cend = cbeg + CHUNK;
        if (cend > s1) cend = s1;

        // Stage this chunk's neighbor indices once per block. Clamped address:
        // every s_idx entry is a valid row index, so the compute loop needs no
        // guards (overhang edges have zero A-bits and contribute nothing).
        {
            int t = cbeg + tid;
            if (t >= s1) t = s1 - 1;           // s1 >= 1 whenever we get here
            s_idx[tid] = __builtin_nontemporal_load(&nidx[t]);
        }
        __syncthreads();

        // Branch-free, EXEC stays all-ones across every WMMA (ISA 7.12).
        // Round trip count up to 4 groups for unrolling; padded groups are
        // harmless (A == 0 there).
        const int eend = cbeg + ((cend - cbeg + 15) & ~15);
#pragma unroll 4
        for (int e = cbeg; e < eend; e += 4) {
            const int ge0 = e + k0;
            const int ge1 = ge0 + 1;

            // A: one-hot membership of edges ge0/ge1 in this lane's row.
            v2f a;
            a.x = ((unsigned)(ge0 - myLo) < myCnt) ? 1.0f : 0.0f;
            a.y = ((unsigned)(ge1 - myLo) < myCnt) ? 1.0f : 0.0f;

            // Two adjacent staged indices -> one LDS access.
            const int2 p = *(const int2*)(s_idx + (ge0 - cbeg));

            // B: gathered features. Keep the POINTER uniform and the whole
            // 32-bit offset per-lane -> saddr-form global_load (GVS mode),
            // one v_lshl_add_u32 per gather instead of 64-bit address math.
            v2f b;
            b.x = feat[((unsigned)p.x << 7) + chan];
            b.y = feat[((unsigned)p.y << 7) + chan];

            // D = A*B + C  (v_wmma_f32_16x16x4_f32, wave32)
            acc = __builtin_amdgcn_wmma_f32_16x16x4_f32(
                /*neg_a=*/false, a, /*neg_b=*/false, b,
                /*c_mod=*/(short)0, acc,
                /*reuse_a=*/false, /*reuse_b=*/false);
        }
        __syncthreads();
    }

    // 1/count per row (mean reduction; empty rows use denom 1).
    if (tid < ROWS_PER_BLOCK) {
        int cnt = s_split[tid + 1] - s_split[tid];
        if (cnt < 1) cnt = 1;
        s_inv[tid] = 1.0f / (float)cnt;
    }
    __syncthreads();

    // D tile layout: VGPR g -> row g (lanes 0-15) / row g+8 (lanes 16-31).
    const int rowOfs = (lane >= 16) ? 8 : 0;
#pragma unroll
    for (int g = 0; g < 8; ++g) {
        const int mrow = g + rowOfs;
        const int m = base + mrow;
        if (m < M) {
            // Streamed output: NT store keeps L2 for the gather working set.
            __builtin_nontemporal_store(acc[g] * s_inv[mrow],
                                        &out[(unsigned)m * C_CHANNELS + chan]);
        }
    }
}

extern "C" void kernel_launch(void* const* d_in, const int* in_sizes, int n_in,
                              void* d_out, int out_size, void* d_ws, size_t ws_size,
                              hipStream_t stream) {
    const float* feat   = (const float*)d_in[0];  // [N*128] f32
    const int*   nidx   = (const int*)  d_in[1];  // [E] int32
    const int*   splits = (const int*)  d_in[2];  // [M+1] int32
    float*       out    = (float*)      d_out;    // [M*128] f32

    const int M = in_sizes[2] - 1;
    const int blocks = (M + ROWS_PER_BLOCK - 1) / ROWS_PER_BLOCK;
    neighbor_pool_wmma<<<blocks, 256, 0, stream>>>(feat, nidx, splits, out, M);
}